// StructuralLoss_4063039062329
// MI455X (gfx1250) — compile-verified
//
#include <hip/hip_runtime.h>
#include <hip/hip_bf16.h>

// ---------------------------------------------------------------------------
// StructuralLoss for MI455X (gfx1250, wave32).
// Bandwidth-bound: ~270MB mandatory traffic -> ~12us floor @ 23.3 TB/s.
// Main pass vectorized (b128 loads, non-temporal pred stream) + fast
// v_exp_f32/v_log_f32 transcendentals to stay under the VALU ceiling.
// WMMA (v_wmma_f32_16x16x4_f32) does the deterministic partial reduction.
// ---------------------------------------------------------------------------

typedef float v2f __attribute__((ext_vector_type(2)));
typedef float v4f __attribute__((ext_vector_type(4)));
typedef float v8f __attribute__((ext_vector_type(8)));

#define PIXELS   16777216   // 16*1024*1024
#define HWSZ     1048576    // 1024*1024
#define NBLK     8192
#define NFIELD   12
#define LOV_N    50000
#define LOV_PAD  65536

// workspace layout helpers (keys first for 8B alignment)
static __host__ __device__ inline unsigned long long* ws_keys(void* ws) {
  return (unsigned long long*)ws;
}
static __host__ __device__ inline float* ws_partials(void* ws) {
  return (float*)((unsigned long long*)ws + 3ull * LOV_PAD);
}
static __host__ __device__ inline float* ws_reduced(void* ws) {
  return ws_partials(ws) + (size_t)NBLK * NFIELD;
}
static __host__ __device__ inline int* ws_flags(void* ws) {      // 64 ints: [0..31] gt, [32..63] pred
  return (int*)(ws_reduced(ws) + 16);
}
static __host__ __device__ inline float* ws_losses(void* ws) {   // 3 floats
  return (float*)(ws_flags(ws) + 64);
}

// ---------------------------------------------------------------------------
__global__ void sl_init_kernel(void* ws) {
  int t = threadIdx.x;
  int* flags = ws_flags(ws);
  float* losses = ws_losses(ws);
  float* red = ws_reduced(ws);
  if (t < 64) flags[t] = 0;
  if (t < 3)  losses[t] = 0.f;
  if (t < 16) red[t] = 0.f;
}

// ---------------------------------------------------------------------------
// Fused main pass: CE, dice sums, boundary, topo flags. 4 pixels per thread
// per iteration (vectorized b128 loads; a quad never crosses a row).
// Fields: 0 ce | 1..3 inter_c | 4..6 psum_c | 7..9 cnt_c | 10 bnum | 11 bden
__global__ void sl_main_kernel(const float* __restrict__ pred,
                               const int* __restrict__ lab,
                               float* __restrict__ partials,
                               int* __restrict__ flags) {
  __shared__ int sgt[32];
  __shared__ int spr[32];
  __shared__ float red[8][NFIELD];
  if (threadIdx.x < 32) { sgt[threadIdx.x] = 0; spr[threadIdx.x] = 0; }
  __syncthreads();

  float acc[NFIELD];
#pragma unroll
  for (int f = 0; f < NFIELD; ++f) acc[f] = 0.f;

  const int stride4 = gridDim.x * blockDim.x * 4;
  for (int p = (blockIdx.x * blockDim.x + threadIdx.x) * 4; p < PIXELS; p += stride4) {
    const int b  = p >> 20;
    const int hw = p & (HWSZ - 1);
    const int h  = hw >> 10;
    const int w  = hw & 1023;          // multiple of 4
    const int base = b * 3 * HWSZ + hw;

    const v4f P0 = __builtin_nontemporal_load((const v4f*)(pred + base));
    const v4f P1 = __builtin_nontemporal_load((const v4f*)(pred + base + HWSZ));
    const v4f P2 = __builtin_nontemporal_load((const v4f*)(pred + base + 2 * HWSZ));
    const int4 LC = *(const int4*)(lab + p);
    int4 LU = make_int4(0, 0, 0, 0);
    int4 LD = make_int4(0, 0, 0, 0);
    if (h > 0)    LU = *(const int4*)(lab + p - 1024);
    if (h < 1023) LD = *(const int4*)(lab + p + 1024);
    const int lL = (w > 0)    ? lab[p - 1] : 0;   // left of element 0
    const int lR = (w < 1020) ? lab[p + 4] : 0;   // right of element 3

    const float a0[4] = {P0[0], P0[1], P0[2], P0[3]};
    const float a1[4] = {P1[0], P1[1], P1[2], P1[3]};
    const float a2[4] = {P2[0], P2[1], P2[2], P2[3]};
    const int lc[4] = {LC.x, LC.y, LC.z, LC.w};
    const int lu[4] = {LU.x, LU.y, LU.z, LU.w};
    const int ld[4] = {LD.x, LD.y, LD.z, LD.w};
    const int ll[4] = {lL, LC.x, LC.y, LC.z};
    const int lr[4] = {LC.y, LC.z, LC.w, lR};

#pragma unroll
    for (int e = 0; e < 4; ++e) {
      const float p0 = a0[e], p1 = a1[e], p2 = a2[e];
      const int l = lc[e];
      const int we = w + e;

      const float m  = fmaxf(p0, fmaxf(p1, p2));
      const float e0 = __expf(p0 - m), e1 = __expf(p1 - m), e2 = __expf(p2 - m);
      const float sum = e0 + e1 + e2;
      const float inv = 1.f / sum;
      const float pr0 = e0 * inv, pr1 = e1 * inv, pr2 = e2 * inv;
      const float lse = m + __logf(sum);
      const float lp  = (l == 0) ? p0 : (l == 1) ? p1 : p2;
      const float ce  = lse - lp;

      acc[0] += ce;
      acc[1] += (l == 0) ? pr0 : 0.f;
      acc[2] += (l == 1) ? pr1 : 0.f;
      acc[3] += (l == 2) ? pr2 : 0.f;
      acc[4] += pr0; acc[5] += pr1; acc[6] += pr2;
      acc[7] += (l == 0) ? 1.f : 0.f;
      acc[8] += (l == 1) ? 1.f : 0.f;
      acc[9] += (l == 2) ? 1.f : 0.f;

      // np.gradient-style boundary mask on classes {1,2}
      bool bnd = false;
#pragma unroll
      for (int cls = 1; cls <= 2; ++cls) {
        const float cc = (l == cls)     ? 1.f : 0.f;
        const float cl = (ll[e] == cls) ? 1.f : 0.f;
        const float cr = (lr[e] == cls) ? 1.f : 0.f;
        const float cu = (lu[e] == cls) ? 1.f : 0.f;
        const float cd = (ld[e] == cls) ? 1.f : 0.f;
        const float gx = (we == 0) ? (cr - cc) : (we == 1023) ? (cc - cl) : 0.5f * (cr - cl);
        const float gy = (h  == 0) ? (cd - cc) : (h  == 1023) ? (cc - cu) : 0.5f * (cd - cu);
        if (fabsf(gx) + fabsf(gy) > 0.1f) bnd = true;
      }
      if (bnd) { acc[10] += ce; acc[11] += 1.f; }

      // topo flags (guarded reads keep LDS atomics rare)
      if (l == 1 && sgt[b * 2]     == 0) atomicOr(&sgt[b * 2], 1);
      if (l == 2 && sgt[b * 2 + 1] == 0) atomicOr(&sgt[b * 2 + 1], 1);
      if (pr1 > 0.5f && spr[b * 2]     == 0) atomicOr(&spr[b * 2], 1);
      if (pr2 > 0.5f && spr[b * 2 + 1] == 0) atomicOr(&spr[b * 2 + 1], 1);
    }
  }

  // wave32 reduce then cross-wave via LDS (deterministic order)
#pragma unroll
  for (int f = 0; f < NFIELD; ++f)
    for (int off = 16; off > 0; off >>= 1)
      acc[f] += __shfl_down(acc[f], off, 32);

  const int wave = threadIdx.x >> 5;
  const int lane = threadIdx.x & 31;
  if (lane == 0)
    for (int f = 0; f < NFIELD; ++f) red[wave][f] = acc[f];
  __syncthreads();

  if (threadIdx.x < NFIELD) {
    float s = 0.f;
    for (int wv = 0; wv < 8; ++wv) s += red[wv][threadIdx.x];
    partials[blockIdx.x * NFIELD + threadIdx.x] = s;
  }

  if (threadIdx.x < 32) {
    if (sgt[threadIdx.x]) atomicOr(&flags[threadIdx.x], 1);
    if (spr[threadIdx.x]) atomicOr(&flags[32 + threadIdx.x], 1);
  }
}

// ---------------------------------------------------------------------------
// WMMA reduce: 12 waves, wave f sums partials[:,f] (8192 values) with
// v_wmma_f32_16x16x4_f32 chained through C (B = ones => row-sum accumulator).
__global__ void sl_reduce_kernel(const float* __restrict__ partials,
                                 float* __restrict__ reduced) {
  const int wave = threadIdx.x >> 5;   // 0..11 = field
  const int lane = threadIdx.x & 31;

  v8f acc = {};
  v2f ones; ones[0] = 1.0f; ones[1] = 1.0f;

  for (int t = 0; t < NBLK / 64; ++t) {
    const int j = t * 64 + lane * 2;
    v2f a;
    a[0] = partials[(size_t)j * NFIELD + wave];
    a[1] = partials[(size_t)(j + 1) * NFIELD + wave];
    // D = A(16x4) * ones(4x16) + C  -> accumulates row sums of all 64 values
    acc = __builtin_amdgcn_wmma_f32_16x16x4_f32(
        /*neg_a=*/false, a, /*neg_b=*/false, ones,
        /*c_mod=*/(short)0, acc, /*reuse_a=*/false, /*reuse_b=*/false);
  }
  // column N=0 lives in lane 0 (M=0..7) and lane 16 (M=8..15)
  float s = acc[0] + acc[1] + acc[2] + acc[3] + acc[4] + acc[5] + acc[6] + acc[7];
  const float t0 = __shfl(s, 0, 32);
  const float t1 = __shfl(s, 16, 32);
  if (lane == 0) reduced[wave] = t0 + t1;
}

// ---------------------------------------------------------------------------
// Lovasz gather: 50000 samples/class via odd-multiplier bijection on 2^24,
// encode u64 sort key: [63:32] descending-order error bits, [31:1] stable
// sample index, [0] fg. Pad to 65536 with max keys.
__global__ void sl_gather_kernel(const float* __restrict__ pred,
                                 const int* __restrict__ lab,
                                 unsigned long long* __restrict__ keys) {
  const int gid = blockIdx.x * blockDim.x + threadIdx.x;  // 0 .. 3*65536-1
  const int c = gid >> 16;
  const int j = gid & (LOV_PAD - 1);
  if (c >= 3) return;
  if (j >= LOV_N) { keys[gid] = ~0ull; return; }

  const unsigned g = ((unsigned)j * 2654435761u) & 0xFFFFFFu;  // distinct pixel idx
  const int b  = (int)(g >> 20);
  const int hw = (int)(g & (HWSZ - 1));
  const int base = b * 3 * HWSZ + hw;

  const float p0 = pred[base];
  const float p1 = pred[base + HWSZ];
  const float p2 = pred[base + 2 * HWSZ];
  const float m  = fmaxf(p0, fmaxf(p1, p2));
  const float e0 = __expf(p0 - m), e1 = __expf(p1 - m), e2 = __expf(p2 - m);
  const float inv = 1.f / (e0 + e1 + e2);
  const float pc = ((c == 0) ? e0 : (c == 1) ? e1 : e2) * inv;

  const int l = lab[g];
  const float fg    = (l == c) ? 1.f : 0.f;
  const float logit = fg - pc;
  const float sign  = 2.f * fg - 1.f;
  const float err   = 1.f - logit * sign;

  const unsigned u = __float_as_uint(err);
  const unsigned s = (u & 0x80000000u) ? ~u : (u | 0x80000000u);  // ascending map
  const unsigned d = ~s;                                          // descending map
  keys[gid] = ((unsigned long long)d << 32) |
              ((unsigned long long)((unsigned)j << 1)) |
              (unsigned long long)((l == c) ? 1u : 0u);
}

// ---------------------------------------------------------------------------
// Bitonic sort, one workgroup per class over its 65536-key segment (L2-resident).
__global__ void sl_sort_kernel(unsigned long long* __restrict__ keys) {
  unsigned long long* base = keys + (size_t)blockIdx.x * LOV_PAD;
  const int n = LOV_PAD;
  for (int k = 2; k <= n; k <<= 1) {
    for (int j = k >> 1; j > 0; j >>= 1) {
      for (int i = threadIdx.x; i < n; i += blockDim.x) {
        const int ij = i ^ j;
        if (ij > i) {
          const unsigned long long a  = base[i];
          const unsigned long long bb = base[ij];
          const bool up = ((i & k) == 0);
          if ((a > bb) == up) { base[i] = bb; base[ij] = a; }
        }
      }
      __syncthreads();
    }
  }
}

// ---------------------------------------------------------------------------
// Lovasz scan+dot: prefix-sum fg, closed-form gradient, relu(err)*grad dot.
__global__ void sl_scan_kernel(const unsigned long long* __restrict__ keys,
                               float* __restrict__ losses) {
  const int c = blockIdx.x;
  const unsigned long long* base = keys + (size_t)c * LOV_PAD;
  const int CH = (LOV_N + 1023) / 1024;  // 49
  const int tid = threadIdx.x;
  const int start = tid * CH;
  const int end = (start + CH < LOV_N) ? (start + CH) : LOV_N;

  float local = 0.f;
  for (int i = start; i < end; ++i) local += (float)(base[i] & 1ull);

  __shared__ float sc[1024];
  sc[tid] = local;
  __syncthreads();
  for (int off = 1; off < 1024; off <<= 1) {
    const float v = (tid >= off) ? sc[tid - off] : 0.f;
    __syncthreads();
    sc[tid] += v;
    __syncthreads();
  }
  const float gts  = sc[1023];
  const float excl = sc[tid] - local;

  float partial = 0.f;
  if (gts > 0.f) {
    float run = excl;
    for (int i = start; i < end; ++i) {
      const unsigned long long kk = base[i];
      const float fg = (float)(kk & 1ull);
      const unsigned d = (unsigned)(kk >> 32);
      const unsigned s = ~d;
      const unsigned u = (s & 0x80000000u) ? (s & 0x7FFFFFFFu) : ~s;
      const float err = __uint_as_float(u);
      run += fg;
      const float jacc = 1.f - (gts - run) / fmaxf(gts + (float)(i + 1) - run, 1e-12f);
      float grad;
      if (i == 0) {
        grad = jacc;
      } else {
        const float rm = run - fg;
        const float jm = 1.f - (gts - rm) / fmaxf(gts + (float)i - rm, 1e-12f);
        grad = jacc - jm;
      }
      partial += fmaxf(err, 0.f) * grad;
    }
  }
  __syncthreads();
  sc[tid] = partial;
  __syncthreads();
  for (int off = 512; off > 0; off >>= 1) {
    if (tid < off) sc[tid] += sc[tid + off];
    __syncthreads();
  }
  if (tid == 0) losses[c] = sc[0];
}

// ---------------------------------------------------------------------------
__global__ void sl_final_kernel(const float* __restrict__ reduced,
                                const int* __restrict__ flags,
                                const float* __restrict__ losses,
                                float* __restrict__ out) {
  if (threadIdx.x != 0 || blockIdx.x != 0) return;
  const float ce = reduced[0] / (float)PIXELS;

  float dsum = 0.f;
  for (int c = 0; c < 3; ++c) {
    const float inter = reduced[1 + c];
    const float denom = reduced[4 + c] + reduced[7 + c];
    dsum += (2.f * inter + 1e-5f) / (denom + 1e-5f);
  }
  const float dice = 1.f - dsum / 3.f;

  float wsum = 0.f, lsum = 0.f;
  for (int c = 0; c < 3; ++c) {
    const float present = (reduced[7 + c] > 0.f) ? 1.f : 0.f;
    wsum += present;
    lsum += losses[c] * present;
  }
  const float lov = lsum / fmaxf(wsum, 1.f);

  const float boundary = reduced[10] / (reduced[11] + 1e-8f);

  float topo = 0.f;
  for (int t = 0; t < 32; ++t) {
    const float g = flags[t] ? 1.f : 0.f;
    const float p = flags[32 + t] ? 1.f : 0.f;
    const float dlt = fmaxf(p - g, 0.f);
    topo += dlt * dlt;
  }
  topo /= 16.f;

  out[0] = 0.4f * ce + 0.3f * dice + 0.2f * lov + 0.08f * boundary + 0.02f * topo;
}

// ---------------------------------------------------------------------------
extern "C" void kernel_launch(void* const* d_in, const int* in_sizes, int n_in,
                              void* d_out, int out_size, void* d_ws, size_t ws_size,
                              hipStream_t stream) {
  const float* pred = (const float*)d_in[0];
  const int*   lab  = (const int*)d_in[1];
  float* out = (float*)d_out;

  unsigned long long* keys = ws_keys(d_ws);
  float* partials = ws_partials(d_ws);
  float* reduced  = ws_reduced(d_ws);
  int*   flags    = ws_flags(d_ws);
  float* losses   = ws_losses(d_ws);

  sl_init_kernel<<<1, 128, 0, stream>>>(d_ws);
  sl_main_kernel<<<NBLK, 256, 0, stream>>>(pred, lab, partials, flags);
  sl_reduce_kernel<<<1, 384, 0, stream>>>(partials, reduced);
  sl_gather_kernel<<<(3 * LOV_PAD) / 256, 256, 0, stream>>>(pred, lab, keys);
  sl_sort_kernel<<<3, 1024, 0, stream>>>(keys);
  sl_scan_kernel<<<3, 1024, 0, stream>>>(keys, losses);
  sl_final_kernel<<<1, 1, 0, stream>>>(reduced, flags, losses, out);
}